// VoxelSetAbstraction_23244363006088
// MI455X (gfx1250) — compile-verified
//
#include <hip/hip_runtime.h>
#include <hip/hip_bf16.h>

// ---------------------------------------------------------------------------
// VoxelSetAbstraction for MI455X (gfx1250, wave32)
//  - bilinear BEV gather: memory/gather bound, plain global loads
//  - keypoint MLP (3->8->16->32 with train-mode BN): BN folded analytically
//    via input moment matrices, matmuls done with V_WMMA_F32_16X16X4_F32
// ---------------------------------------------------------------------------

typedef __attribute__((ext_vector_type(2))) float v2f;
typedef __attribute__((ext_vector_type(8))) float v8f;

#define NROWS 16384   // B*K
#define KPTS  4096
#define CBEV  128
#define HH    400
#define WW    400
#define BN_EPS 1e-5f

// ---------------- zero scratch (stats accumulators) ----------------
__global__ void zero_ws_kernel(float* p, int n) {
  int i = blockIdx.x * blockDim.x + threadIdx.x;
  if (i < n) p[i] = 0.f;
}

// ---------------- bilinear BEV feature gather ----------------
// 2 keypoints per 256-thread block; thread = (kp, channel)
__global__ void bilinear_kernel(const float* __restrict__ kp,
                                const float* __restrict__ feat,
                                const int* __restrict__ stride_p,
                                float* __restrict__ out) {
  int t = threadIdx.x;
  int kpi = blockIdx.x * 2 + (t >> 7);
  int c = t & 127;
  float stride = (float)stride_p[0];
  float inv = 1.0f / (0.05f * stride);   // 1 / (VOXEL * stride)
  float kx = kp[(size_t)kpi * 4 + 0];
  float ky = kp[(size_t)kpi * 4 + 1];
  float x = (kx - 0.0f) * inv;           // PC_MIN_X = 0
  float y = (ky + 40.0f) * inv;          // PC_MIN_Y = -40
  float xf = floorf(x), yf = floorf(y);
  int x0 = (int)xf, y0 = (int)yf;
  int x1 = x0 + 1, y1 = y0 + 1;
  x0 = min(max(x0, 0), WW - 1); x1 = min(max(x1, 0), WW - 1);
  y0 = min(max(y0, 0), HH - 1); y1 = min(max(y1, 0), HH - 1);
  float x0f = (float)x0, x1f = (float)x1, y0f = (float)y0, y1f = (float)y1;
  float wa = (x1f - x) * (y1f - y);
  float wb = (x1f - x) * (y - y0f);
  float wc = (x - x0f) * (y1f - y);
  float wd = (x - x0f) * (y - y0f);
  int b = kpi / KPTS;
  const float* plane = feat + ((size_t)(b * CBEV + c)) * HH * WW;
  float Ia = plane[y0 * WW + x0];
  float Ib = plane[y1 * WW + x0];
  float Ic = plane[y0 * WW + x1];
  float Id = plane[y1 * WW + x1];
  out[(size_t)kpi * 160 + c] = Ia * wa + Ib * wb + Ic * wc + Id * wd;
}

// ---------------- column sums + second-moment matrix of an (nrows x NF) matrix ----------------
// acc layout: [0..NF) = column sums, [NF..NF+NF*NF) = sum of x x^T
template <int NF>
__global__ void stats_kernel(const float* __restrict__ in, int in_stride,
                             int nrows, float* __restrict__ acc) {
  const int TILE = 16;
  __shared__ float tile[TILE][NF];
  int t = threadIdx.x;
  int mi = (t < NF * NF) ? (t / NF) : 0;
  int mj = (t < NF * NF) ? (t % NF) : 0;
  float lsum = 0.f, lmom = 0.f;
  int tiles_total = nrows / TILE;
  for (int tb = blockIdx.x; tb < tiles_total; tb += gridDim.x) {
    int rowbase = tb * TILE;
    for (int idx = t; idx < TILE * NF; idx += blockDim.x)
      tile[idx / NF][idx % NF] =
          in[(size_t)(rowbase + idx / NF) * in_stride + (idx % NF)];
    __syncthreads();
    if (t < NF * NF) {
      #pragma unroll
      for (int r = 0; r < TILE; ++r) lmom += tile[r][mi] * tile[r][mj];
    }
    if (t < NF) {
      #pragma unroll
      for (int r = 0; r < TILE; ++r) lsum += tile[r][t];
    }
    __syncthreads();
  }
  if (t < NF * NF) atomicAdd(&acc[NF + t], lmom);
  if (t < NF) atomicAdd(&acc[t], lsum);
}

// ---------------- fold BN(train) into effective weight/bias ----------------
// z = x@W ; mean_j = mu_x . W_:j ; E[z_j^2] = W_:j^T (Mom/N) W_:j
// h = relu(x @ (W*a) + (b - mean*a)), a = g * rsqrt(var + eps)
template <int KIN, int NOUT, int KPAD, int NPAD>
__global__ void fold_bn_kernel(const float* __restrict__ acc,
                               const float* __restrict__ W,
                               const float* __restrict__ g,
                               const float* __restrict__ bb,
                               float* __restrict__ weff,
                               float* __restrict__ beff, float invN) {
  int j = threadIdx.x;
  if (j >= NPAD) return;
  float a = 0.f, be = 0.f;
  if (j < NOUT) {
    float mean = 0.f;
    for (int i = 0; i < KIN; ++i) mean += acc[i] * W[i * NOUT + j];
    mean *= invN;
    float ex2 = 0.f;
    for (int i = 0; i < KIN; ++i) {
      float wi = W[i * NOUT + j];
      for (int k = 0; k < KIN; ++k)
        ex2 += wi * W[k * NOUT + j] * acc[KIN + i * KIN + k];
    }
    ex2 *= invN;
    float var = ex2 - mean * mean;
    a = g[j] * rsqrtf(var + BN_EPS);
    be = bb[j] - mean * a;
  }
  for (int k = 0; k < KPAD; ++k)
    weff[k * NPAD + j] = (k < KIN && j < NOUT) ? W[k * NOUT + j] * a : 0.f;
  beff[j] = be;
}

// ---------------- relu(x @ Weff + beff) via V_WMMA_F32_16X16X4_F32 ----------------
// 8 waves/block, each wave owns a 16-row tile; K chained in chunks of 4.
// A 16x4 f32 layout: lanes 0-15: K=0 (v0), K=1 (v1); lanes 16-31: K=2, K=3.
// C/D layout: VGPR r: lanes 0-15 -> M=r, lanes 16-31 -> M=r+8; N = lane&15.
template <int KPAD, int NPAD>
__global__ void mlp_wmma_kernel(const float* __restrict__ in, int in_stride,
                                int k_real,
                                const float* __restrict__ weff,
                                const float* __restrict__ beff,
                                float* __restrict__ out, int out_stride,
                                int n_real, int nrows) {
  int lane = threadIdx.x & 31;
  int wid = threadIdx.x >> 5;
  int base = (blockIdx.x * 8 + wid) * 16;
  if (base >= nrows) return;
  int hi = lane >> 4;     // 0: lanes 0-15, 1: lanes 16-31
  int ln = lane & 15;
  int rowA = base + ln;
  #pragma unroll
  for (int nt = 0; nt < NPAD / 16; ++nt) {
    int col = ln + nt * 16;
    float bv = beff[col];
    v8f c;
    #pragma unroll
    for (int r = 0; r < 8; ++r) c[r] = bv;   // C initialized with folded bias
    #pragma unroll
    for (int kk = 0; kk < KPAD; kk += 4) {
      int kb = kk + hi * 2;
      v2f a, b;
      a.x = (kb + 0 < k_real) ? in[(size_t)rowA * in_stride + kb + 0] : 0.f;
      a.y = (kb + 1 < k_real) ? in[(size_t)rowA * in_stride + kb + 1] : 0.f;
      b.x = weff[(kb + 0) * NPAD + col];     // weff is zero-padded to KPAD x NPAD
      b.y = weff[(kb + 1) * NPAD + col];
      c = __builtin_amdgcn_wmma_f32_16x16x4_f32(false, a, false, b,
                                                (short)0, c, false, false);
    }
    #pragma unroll
    for (int r = 0; r < 8; ++r) {
      float v = c[r] > 0.f ? c[r] : 0.f;     // ReLU
      int row = base + r + hi * 8;
      if (col < n_real) out[(size_t)row * out_stride + col] = v;
    }
  }
}

// ---------------------------------------------------------------------------
extern "C" void kernel_launch(void* const* d_in, const int* in_sizes, int n_in,
                              void* d_out, int out_size, void* d_ws,
                              size_t ws_size, hipStream_t stream) {
  const float* keypoints = (const float*)d_in[0];
  const float* spatial   = (const float*)d_in[1];
  const float* w1 = (const float*)d_in[2];
  const float* g1 = (const float*)d_in[3];
  const float* b1 = (const float*)d_in[4];
  const float* w2 = (const float*)d_in[5];
  const float* g2 = (const float*)d_in[6];
  const float* b2 = (const float*)d_in[7];
  const float* w3 = (const float*)d_in[8];
  const float* g3 = (const float*)d_in[9];
  const float* b3 = (const float*)d_in[10];
  const int* stride_p = (const int*)d_in[11];
  float* out = (float*)d_out;
  float* ws = (float*)d_ws;

  // workspace layout (floats)
  float* stats0 = ws + 0;     // 3 + 9
  float* stats1 = ws + 16;    // 8 + 64
  float* stats2 = ws + 96;    // 16 + 256
  float* w1eff = ws + 384;  float* b1eff = w1eff + 4 * 16;
  float* w2eff = ws + 480;  float* b2eff = w2eff + 8 * 16;
  float* w3eff = ws + 640;  float* b3eff = w3eff + 16 * 32;
  float* h1 = ws + 1200;               // 16384 x 8
  float* h2 = h1 + (size_t)NROWS * 8;  // 16384 x 16

  const float invN = 1.0f / (float)NROWS;

  zero_ws_kernel<<<2, 256, 0, stream>>>(ws, 384);

  // BEV bilinear features -> out[:, 0:128]
  bilinear_kernel<<<NROWS / 2, 256, 0, stream>>>(keypoints, spatial, stride_p, out);

  // layer 1: xyz(3) -> 8
  stats_kernel<3><<<64, 256, 0, stream>>>(keypoints, 4, NROWS, stats0);
  fold_bn_kernel<3, 8, 4, 16><<<1, 32, 0, stream>>>(stats0, w1, g1, b1, w1eff, b1eff, invN);
  mlp_wmma_kernel<4, 16><<<NROWS / 128, 256, 0, stream>>>(
      keypoints, 4, 3, w1eff, b1eff, h1, 8, 8, NROWS);

  // layer 2: 8 -> 16
  stats_kernel<8><<<64, 256, 0, stream>>>(h1, 8, NROWS, stats1);
  fold_bn_kernel<8, 16, 8, 16><<<1, 32, 0, stream>>>(stats1, w2, g2, b2, w2eff, b2eff, invN);
  mlp_wmma_kernel<8, 16><<<NROWS / 128, 256, 0, stream>>>(
      h1, 8, 8, w2eff, b2eff, h2, 16, 16, NROWS);

  // layer 3: 16 -> 32, written straight into out[:, 128:160]
  stats_kernel<16><<<64, 256, 0, stream>>>(h2, 16, NROWS, stats2);
  fold_bn_kernel<16, 32, 16, 32><<<1, 32, 0, stream>>>(stats2, w3, g3, b3, w3eff, b3eff, invN);
  mlp_wmma_kernel<16, 32><<<NROWS / 128, 256, 0, stream>>>(
      h2, 16, 16, w3eff, b3eff, out + 128, 160, 32, NROWS);
}